// MIND_2396591751877
// MI455X (gfx1250) — compile-verified
//
#include <hip/hip_runtime.h>
#include <hip/hip_bf16.h>

typedef __attribute__((ext_vector_type(2))) float v2f;
typedef __attribute__((ext_vector_type(4))) float v4f;
typedef __attribute__((ext_vector_type(8))) float v8f;
typedef __attribute__((ext_vector_type(4))) int   i32x4;

// address-space-qualified int4 for the async-to-LDS builtin
typedef __attribute__((address_space(1))) i32x4 gas_i32x4;   // global
typedef __attribute__((address_space(3))) i32x4 las_i32x4;   // LDS

#define NB   1024   // batch
#define NL   512    // sequence length
#define ND   64     // D_in == D_out
#define NK   8      // capsules
#define NEGV (-65535.0f)

// ---------------------------------------------------------------------------
// Async global -> LDS (gfx1250 GLOBAL_LOAD_ASYNC_TO_LDS_B128, ASYNCcnt),
// with a register-copy fallback if the builtin is unavailable.
// ---------------------------------------------------------------------------
#if defined(__has_builtin)
#if __has_builtin(__builtin_amdgcn_global_load_async_to_lds_b128)
#define HAVE_ASYNC 1
#endif
#endif
#ifndef HAVE_ASYNC
#define HAVE_ASYNC 0
#endif

__device__ __forceinline__ void g2l_b128(const float* __restrict__ g, float* l) {
#if HAVE_ASYNC
  __builtin_amdgcn_global_load_async_to_lds_b128(
      (gas_i32x4*)g, (las_i32x4*)l, 0, 0);
#else
  *(v4f*)l = *(const v4f*)g;
#endif
}

template <int N>
__device__ __forceinline__ void wait_async_le() {
#if HAVE_ASYNC
#if __has_builtin(__builtin_amdgcn_s_wait_asynccnt)
  __builtin_amdgcn_s_wait_asynccnt((unsigned short)N);
#else
  asm volatile("s_wait_asynccnt %0" ::"i"(N) : "memory");
#endif
#endif
}

// ---------------------------------------------------------------------------
// low_new = low_capsule @ S   ([B*L,64] x [64,64] -> [B*L,64]), f32 WMMA
// grid: (B*L)/64 blocks, 128 threads (4 waves); each wave owns a 16x64 strip.
// ---------------------------------------------------------------------------
__global__ __launch_bounds__(128) void proj_kernel(
    const float* __restrict__ X, const float* __restrict__ S,
    float* __restrict__ Y) {
  __shared__ float s_S[64 * 80];      // pitch 80: halves of wave hit disjoint banks
  __shared__ float s_A[4][16 * 68];   // pitch 68: 4*m mod 64 distinct for m=0..15

  const int tid  = threadIdx.x;
  const int lane = tid & 31;
  const int w    = tid >> 5;
  const int g    = lane >> 4;   // half-wave select (K sub-pair)
  const int m    = lane & 15;   // row / col within tile

  // async-stage S into LDS (padded)
#pragma unroll
  for (int i = 0; i < 8; ++i) {
    int fi = i * 512 + tid * 4;               // 4096 floats total
    g2l_b128(S + fi, &s_S[(fi >> 6) * 80 + (fi & 63)]);
  }
  // async-stage this wave's 16x64 A strip (coalesced)
  const long rowbase = (long)blockIdx.x * 64 + w * 16;
#pragma unroll
  for (int it = 0; it < 8; ++it) {
    int flat = it * 128 + lane * 4;           // 0..1023
    g2l_b128(X + rowbase * 64 + flat, &s_A[w][(flat >> 6) * 68 + (flat & 63)]);
  }
  wait_async_le<0>();
  __syncthreads();

  // hoist A fragments: afr[s] = A[m][4s+2g .. +1]
  v2f afr[16];
#pragma unroll
  for (int s = 0; s < 16; ++s) {
    int c = s * 4 + 2 * g;
    afr[s].x = s_A[w][m * 68 + c];
    afr[s].y = s_A[w][m * 68 + c + 1];
  }

#pragma unroll
  for (int nt = 0; nt < 4; ++nt) {
    v8f acc = {};
#pragma unroll
    for (int s = 0; s < 16; ++s) {
      int c = s * 4 + 2 * g;
      v2f bf;
      bf.x = s_S[c * 80 + nt * 16 + m];
      bf.y = s_S[(c + 1) * 80 + nt * 16 + m];
      acc = __builtin_amdgcn_wmma_f32_16x16x4_f32(
          false, afr[s], false, bf, (short)0, acc, false, false);
    }
#pragma unroll
    for (int r = 0; r < 8; ++r) {
      long row = rowbase + r + 8 * g;         // D layout: vgpr r -> M = r + 8*g
      Y[row * 64 + nt * 16 + m] = acc[r];
    }
  }
}

// ---------------------------------------------------------------------------
// One routing iteration for one batch element per block.
// block = 128 threads (4 waves). low_new is streamed through LDS with a
// double-buffered async pipeline (8 chunks x 64 rows). Writes high -> out and
// per-batch B_delta partials -> partial[b][k][l] (reduced later, fixed order).
// ---------------------------------------------------------------------------
__device__ __forceinline__ void stage_chunk(const float* __restrict__ src,
                                            float* dst, int tid) {
#pragma unroll
  for (int i = 0; i < 8; ++i) {
    int flat = i * 512 + tid * 4;             // 64 rows x 64 floats
    g2l_b128(src + flat, dst + (flat >> 6) * 72 + (flat & 63));
  }
}

__global__ __launch_bounds__(128) void route_kernel(
    const float* __restrict__ low_new, const float* __restrict__ Bmat,
    const int* __restrict__ seq_len, float* __restrict__ out_high,
    float* __restrict__ partial) {
  __shared__ float s_chunk[2][64 * 72];   // pitch 72: halves hit disjoint banks
  __shared__ float s_W[NK * 521];         // softmax weights, conflict-free pitch
  __shared__ float s_H[NK * 68];          // squashed high tile

  const int b    = blockIdx.x;
  const int tid  = threadIdx.x;
  const int lane = tid & 31;
  const int w    = tid >> 5;
  const int g    = lane >> 4;
  const int m    = lane & 15;
  const int sl   = seq_len[b];

  const float* ln = low_new + (long)b * NL * ND;

  // kick off chunk 0 of low_new while we do the softmax
  stage_chunk(ln, s_chunk[0], tid);

  // --- masked softmax over L for rows k = 2w, 2w+1 (one wave per 2 rows) ---
  for (int kr = 2 * w; kr < 2 * w + 2; ++kr) {
    float vals[16];
    float mx = -3.0e38f;
#pragma unroll
    for (int i = 0; i < 16; ++i) {
      int l = lane + 32 * i;
      float v = (l < sl) ? Bmat[kr * NL + l] : NEGV;
      vals[i] = v;
      mx = fmaxf(mx, v);
    }
#pragma unroll
    for (int o = 16; o > 0; o >>= 1) mx = fmaxf(mx, __shfl_xor(mx, o, 32));
    float ssum = 0.0f;
#pragma unroll
    for (int i = 0; i < 16; ++i) {
      float e = __expf(vals[i] - mx);
      ssum += e;
      vals[i] = e;
    }
#pragma unroll
    for (int o = 16; o > 0; o >>= 1) ssum += __shfl_xor(ssum, o, 32);
    float rs = 1.0f / ssum;
#pragma unroll
    for (int i = 0; i < 16; ++i) s_W[kr * 521 + lane + 32 * i] = vals[i] * rs;
  }

  const float sel = (m < NK) ? 1.0f : 0.0f;   // zero-pad A rows 8..15
  const int   mr  = m & 7;
  const int   o   = w * 16 + m;               // this wave's D_out column

  // --- GEMM1: high = W @ low_new  (M=16 pad, N=16/wave, K=512) ---
  // double-buffered async pipeline over 8 chunks of 64 rows
  v8f acc0 = {}, acc1 = {};
  for (int ch = 0; ch < 8; ++ch) {
    if (ch < 7) {
      stage_chunk(ln + (ch + 1) * 64 * ND, s_chunk[(ch + 1) & 1], tid);
      wait_async_le<8>();                     // retire chunk ch, keep ch+1 in flight
    } else {
      wait_async_le<0>();
    }
    __syncthreads();                          // all waves' async data visible
    const float* cb = s_chunk[ch & 1];
#pragma unroll
    for (int kk = 0; kk < 64; kk += 8) {
      {
        int cw = ch * 64 + kk + 2 * g;        // W column (global L index)
        int cr = kk + 2 * g;                  // row within chunk
        v2f a, bf;
        a.x  = sel * s_W[mr * 521 + cw];
        a.y  = sel * s_W[mr * 521 + cw + 1];
        bf.x = cb[cr * 72 + o];
        bf.y = cb[(cr + 1) * 72 + o];
        acc0 = __builtin_amdgcn_wmma_f32_16x16x4_f32(
            false, a, false, bf, (short)0, acc0, false, false);
      }
      {
        int cw = ch * 64 + kk + 4 + 2 * g;
        int cr = kk + 4 + 2 * g;
        v2f a, bf;
        a.x  = sel * s_W[mr * 521 + cw];
        a.y  = sel * s_W[mr * 521 + cw + 1];
        bf.x = cb[cr * 72 + o];
        bf.y = cb[(cr + 1) * 72 + o];
        acc1 = __builtin_amdgcn_wmma_f32_16x16x4_f32(
            false, a, false, bf, (short)0, acc1, false, false);
      }
    }
    __syncthreads();                          // protect buffer before restage
  }
  v8f acc = acc0 + acc1;

  // --- squash: sum over K lives exactly in this lane's 8 accumulators ---
  float sq = 0.0f;
#pragma unroll
  for (int r = 0; r < 8; ++r) sq += acc[r] * acc[r];
  float scale = sq / (1.0f + sq) * __frsqrt_rn(sq + 1e-9f);
#pragma unroll
  for (int r = 0; r < 8; ++r) acc[r] *= scale;

  if (g == 0) {   // lanes 0..15 hold valid rows k=0..7
#pragma unroll
    for (int r = 0; r < 8; ++r) {
      out_high[((long)b * NK + r) * ND + o] = acc[r];
      s_H[r * 68 + o] = acc[r];
    }
  }
  __syncthreads();

  // --- GEMM2: B_delta[k][l] = high @ low_new^T  (M=16 pad, N=128/wave, K=64) ---
  // low_new rows are L2-resident from GEMM1's pass; direct float2 loads.
  v2f hfr[16];
#pragma unroll
  for (int s = 0; s < 16; ++s) {
    int c = s * 4 + 2 * g;
    hfr[s].x = sel * s_H[mr * 68 + c];
    hfr[s].y = sel * s_H[mr * 68 + c + 1];
  }
  for (int nt = 0; nt < 8; ++nt) {
    int l0 = w * 128 + nt * 16 + m;            // L index (B-matrix column)
    const float* lrow = ln + l0 * ND;
    v8f acc2 = {};
#pragma unroll
    for (int s = 0; s < 16; ++s) {
      int c = s * 4 + 2 * g;
      v2f bf = *(const v2f*)(lrow + c);        // o-pairs contiguous in memory
      acc2 = __builtin_amdgcn_wmma_f32_16x16x4_f32(
          false, hfr[s], false, bf, (short)0, acc2, false, false);
    }
    if (g == 0) {
#pragma unroll
      for (int r = 0; r < 8; ++r)
        partial[((long)b * NK + r) * NL + l0] = acc2[r];
    }
  }
}

// ---------------------------------------------------------------------------
// Fixed-order reduction: Bmat[k][l] += sum_b partial[b][k][l]   (deterministic)
// ---------------------------------------------------------------------------
__global__ __launch_bounds__(256) void reduce_kernel(
    const float* __restrict__ partial, float* __restrict__ Bmat) {
  int t = blockIdx.x * blockDim.x + threadIdx.x;   // 0..4095
  float s = 0.0f;
  for (int b = 0; b < NB; ++b) s += partial[(long)b * (NK * NL) + t];
  Bmat[t] += s;
}

extern "C" void kernel_launch(void* const* d_in, const int* in_sizes, int n_in,
                              void* d_out, int out_size, void* d_ws, size_t ws_size,
                              hipStream_t stream) {
  const float* low_capsule = (const float*)d_in[0];  // [B, L, D]
  const float* B_init      = (const float*)d_in[1];  // [1, K, L]
  const float* S           = (const float*)d_in[2];  // [D, D]
  const int*   seq_len     = (const int*)d_in[3];    // [B, 1]
  float* out = (float*)d_out;                        // [B, K, D]

  char*  ws      = (char*)d_ws;
  float* low_new = (float*)ws;                                   // B*L*D f32
  float* Bmat    = (float*)(ws + (size_t)NB * NL * ND * 4);      // K*L f32
  float* partial = Bmat + NK * NL;                               // B*K*L f32

  (void)hipMemcpyAsync(Bmat, B_init, (size_t)NK * NL * sizeof(float),
                       hipMemcpyDeviceToDevice, stream);

  proj_kernel<<<(NB * NL) / 64, 128, 0, stream>>>(low_capsule, S, low_new);

  for (int it = 0; it < 3; ++it) {
    route_kernel<<<NB, 128, 0, stream>>>(low_new, Bmat, seq_len, out, partial);
    if (it < 2) reduce_kernel<<<(NK * NL) / 256, 256, 0, stream>>>(partial, Bmat);
  }
}